// GNNDecoder_11871289606582
// MI455X (gfx1250) — compile-verified
//
#include <hip/hip_runtime.h>
#include <hip/hip_bf16.h>

#define N_NODES 100000
#define N_EDGES 1600000
#define D 64
#define ROW_TILES (N_NODES / 16)   // 6250, N_NODES % 16 == 0

typedef __attribute__((ext_vector_type(2))) float v2f;
typedef __attribute__((ext_vector_type(8))) float v8f;

// ---------------------------------------------------------------------------
// Degree accumulation: deg[dst[e]] += 1  (float atomics; counts < 2^24 exact)
// ---------------------------------------------------------------------------
__global__ __launch_bounds__(256) void gcn_deg_kernel(
    const int* __restrict__ dst, float* __restrict__ deg)
{
    int e = blockIdx.x * 256 + threadIdx.x;
    if (e < N_EDGES) {
        unsafeAtomicAdd(&deg[dst[e]], 1.0f);
    }
}

// dinv[i] = rsqrt(deg[i] + 1)   (in-place over the deg buffer)
__global__ __launch_bounds__(256) void gcn_dinv_kernel(float* __restrict__ dinv)
{
    int i = blockIdx.x * 256 + threadIdx.x;
    if (i < N_NODES) {
        dinv[i] = rsqrtf(dinv[i] + 1.0f);
    }
}

// ---------------------------------------------------------------------------
// WMMA GEMM: h = (relu?)(x) @ W ; agg = h * dinv[row]^2 + b   (self-loop+bias)
// One wave = 16 rows x 64 cols (4 tiles of 16x16, K=64 via 16x V_WMMA f32 k4).
// Safe to alias x and agg: each wave exclusively owns its 16 rows.
// ---------------------------------------------------------------------------
__global__ __launch_bounds__(256) void gcn_gemm_kernel(
    const float* __restrict__ x, const float* __restrict__ Wg,
    const float* __restrict__ bg, const float* __restrict__ dinv,
    float* __restrict__ h, float* __restrict__ agg, int relu_input)
{
    __shared__ float Ws[D * D];
    __shared__ float bs[D];
    const int tid = threadIdx.x;
    #pragma unroll
    for (int i = 0; i < (D * D) / 256; ++i)
        Ws[tid + i * 256] = Wg[tid + i * 256];
    if (tid < D) bs[tid] = bg[tid];
    __syncthreads();

    const int wave = tid >> 5;
    const int lane = tid & 31;
    const int tile = blockIdx.x * 8 + wave;
    if (tile >= ROW_TILES) return;

    const int row_base = tile * 16;
    const int half = lane >> 4;     // 0: lanes 0-15, 1: lanes 16-31
    const int l15  = lane & 15;

    const float* xrow = x + (size_t)(row_base + l15) * D;

    v8f acc[4] = {v8f{0}, v8f{0}, v8f{0}, v8f{0}};

    #pragma unroll
    for (int k = 0; k < D; k += 4) {
        // A fragment (16x4 f32): lanes 0-15 -> K = k,k+1 ; lanes 16-31 -> K = k+2,k+3
        const int ka = k + 2 * half;
        v2f a;
        a.x = xrow[ka];
        a.y = xrow[ka + 1];
        if (relu_input) {
            a.x = fmaxf(a.x, 0.0f);
            a.y = fmaxf(a.y, 0.0f);
        }
        #pragma unroll
        for (int c = 0; c < 4; ++c) {
            // B fragment (4x16 f32): N = l15 within the tile, same K split as A
            v2f b;
            b.x = Ws[ka * D       + c * 16 + l15];
            b.y = Ws[(ka + 1) * D + c * 16 + l15];
            acc[c] = __builtin_amdgcn_wmma_f32_16x16x4_f32(
                false, a, false, b, (short)0, acc[c], false, false);
        }
    }

    // C/D layout: lanes 0-15 -> rows M = r (VGPR r), lanes 16-31 -> rows M = 8+r; N = l15
    #pragma unroll
    for (int r = 0; r < 8; ++r) {
        const int row = row_base + 8 * half + r;
        const float di  = dinv[row];
        const float di2 = di * di;
        #pragma unroll
        for (int c = 0; c < 4; ++c) {
            const int col = c * 16 + l15;
            const float v = acc[c][r];
            h[(size_t)row * D + col]   = v;
            agg[(size_t)row * D + col] = fmaf(v, di2, bs[col]);
        }
    }
}

// ---------------------------------------------------------------------------
// Edge scatter: agg[dst] += dinv[src]*dinv[dst] * h[src]
// One wave per edge, 2 floats per lane; h/agg are L2-resident (25.6 MB each).
// ---------------------------------------------------------------------------
__global__ __launch_bounds__(256) void gcn_edge_kernel(
    const int* __restrict__ src, const int* __restrict__ dst,
    const float* __restrict__ dinv, const float* __restrict__ h,
    float* __restrict__ agg)
{
    const int wave = threadIdx.x >> 5;
    const int lane = threadIdx.x & 31;
    const long e = (long)blockIdx.x * 8 + wave;
    if (e >= N_EDGES) return;

    const int s = src[e];
    const int t = dst[e];
    const float w = dinv[s] * dinv[t];

    const float2* hv = (const float2*)(h + (size_t)s * D);
    const float2 v = hv[lane];

    float* out = agg + (size_t)t * D + lane * 2;
    unsafeAtomicAdd(out,     v.x * w);
    unsafeAtomicAdd(out + 1, v.y * w);
}

// ---------------------------------------------------------------------------
extern "C" void kernel_launch(void* const* d_in, const int* in_sizes, int n_in,
                              void* d_out, int out_size, void* d_ws, size_t ws_size,
                              hipStream_t stream) {
    const float* z   = (const float*)d_in[0];
    const int*   src = (const int*)  d_in[1];
    const int*   dst = (const int*)  d_in[2];
    const float* W1  = (const float*)d_in[3];
    const float* b1  = (const float*)d_in[4];
    const float* W2  = (const float*)d_in[5];
    const float* b2  = (const float*)d_in[6];
    const float* W3  = (const float*)d_in[7];
    const float* b3  = (const float*)d_in[8];
    float* out = (float*)d_out;

    // Workspace layout: dinv[N] | h[N*64] | agg[N*64]
    float* dinv = (float*)d_ws;
    float* h    = dinv + N_NODES;
    float* agg  = h + (size_t)N_NODES * D;

    const dim3 blk(256);
    const dim3 grid_edges((N_EDGES + 7) / 8);         // one wave per edge
    const dim3 grid_edges_flat((N_EDGES + 255) / 256);
    const dim3 grid_nodes((N_NODES + 255) / 256);
    const dim3 grid_gemm((ROW_TILES + 7) / 8);        // one wave per 16-row tile

    // --- degree / dinv (graph-constant across all three layers) ---
    hipMemsetAsync(dinv, 0, (size_t)N_NODES * sizeof(float), stream);
    gcn_deg_kernel<<<grid_edges_flat, blk, 0, stream>>>(dst, dinv);
    gcn_dinv_kernel<<<grid_nodes, blk, 0, stream>>>(dinv);

    // --- layer 1: x = z (no input relu) ---
    gcn_gemm_kernel<<<grid_gemm, blk, 0, stream>>>(z, W1, b1, dinv, h, agg, 0);
    gcn_edge_kernel<<<grid_edges, blk, 0, stream>>>(src, dst, dinv, h, agg);

    // --- layer 2: x = relu(agg), in-place agg init ---
    gcn_gemm_kernel<<<grid_gemm, blk, 0, stream>>>(agg, W2, b2, dinv, h, agg, 1);
    gcn_edge_kernel<<<grid_edges, blk, 0, stream>>>(src, dst, dinv, h, agg);

    // --- layer 3: x = relu(agg), final agg goes straight to d_out ---
    gcn_gemm_kernel<<<grid_gemm, blk, 0, stream>>>(agg, W3, b3, dinv, h, out, 1);
    gcn_edge_kernel<<<grid_edges, blk, 0, stream>>>(src, dst, dinv, h, out);
}